// SSDTargetTransform_86861418594864
// MI455X (gfx1250) — compile-verified
//
#include <hip/hip_runtime.h>

// ---------------------------------------------------------------------------
// SSD target transform for MI455X (gfx1250).
// B=128 batches, G=64 gt boxes, P=8732 priors.
// K1: column argmax (best prior per gt) via LDS ds_max_u64 + global_atomic_max_u64.
// K2: row argmax per prior (recompute), force-assign, box encode, write out.
// gt tile (1 KB) is DMA'd to LDS with the CDNA5 Tensor Data Mover.
// ---------------------------------------------------------------------------

#define CENTER_VARIANCE 0.1f
#define SIZE_VARIANCE   0.2f
#define IOU_THRESHOLD   0.5f
#define EPS_F           1e-5f

static constexpr int Bn = 128;
static constexpr int Gn = 64;
static constexpr int Pn = 8732;

typedef unsigned int u32x4 __attribute__((ext_vector_type(4)));
typedef int          i32x8 __attribute__((ext_vector_type(8)));
typedef int          i32x4 __attribute__((ext_vector_type(4)));

// --- Tensor Data Mover: DMA a contiguous 1 KB tile (64 boxes * 4 f32) -> LDS.
// D# built per CDNA5 ISA ch.8: group0 = {count/flags, lds_addr, global_addr, type=2},
// group1 = {data_size=4B, tensor_dim0=256, tile_dim0=256, stride=256}, 1-D tile.
__device__ __forceinline__ void tdm_load_gt_tile(const void* gsrc, void* lds_dst) {
  unsigned long long ga = (unsigned long long)(size_t)gsrc;
  unsigned int lds_off  = (unsigned int)(size_t)lds_dst;  // low 32 bits = LDS offset

  u32x4 g0;
  g0.x = 1u;                                             // count=1, user mode, no gather
  g0.y = lds_off;                                        // lds_addr [63:32]
  g0.z = (unsigned int)ga;                               // global_addr [95:64]
  g0.w = (unsigned int)((ga >> 32) & 0x01FFFFFFull)      // global_addr [120:96]
       | (2u << 30);                                     // type=2 ("image")

  i32x8 g1;
  g1[0] = 0x00020000;                   // workgroup_mask=0, data_size=2 (4 bytes)
  g1[1] = (int)(256u << 16);            // tensor_dim0[15:0] = 256 elements
  g1[2] = (int)(1u << 16);              // tensor_dim0 hi=0 | tensor_dim1 lo = 1
  g1[3] = (int)(256u << 16);            // tensor_dim1 hi=0 | tile_dim0 = 256
  g1[4] = 1;                            // tile_dim1 = 1, tile_dim2 = 0
  g1[5] = 256;                          // tensor_dim0_stride[31:0] = 256
  g1[6] = (int)(256u << 16);            // stride hi=0 | tensor_dim1_stride lo = 256
  g1[7] = 0;

  i32x4 z4 = {0, 0, 0, 0};              // groups 2/3 unused (<=2D tensor)
#if defined(__clang_major__) && (__clang_major__ >= 23)
  i32x8 z8 = {0, 0, 0, 0, 0, 0, 0, 0};
  __builtin_amdgcn_tensor_load_to_lds(g0, g1, z4, z4, z8, 0);
#else
  __builtin_amdgcn_tensor_load_to_lds(g0, g1, z4, z4, 0);
#endif
}

__device__ __forceinline__ float iou_one(float px1, float py1, float px2, float py2,
                                         float pa, float4 gb, float ga) {
  float ix1 = fmaxf(px1, gb.x), iy1 = fmaxf(py1, gb.y);
  float ix2 = fminf(px2, gb.z), iy2 = fminf(py2, gb.w);
  float iw = fmaxf(ix2 - ix1, 0.0f), ih = fmaxf(iy2 - iy1, 0.0f);
  float inter = iw * ih;
  return inter / (pa + ga - inter + EPS_F);
}

// K0: zero the column-max workspace (d_ws is poisoned; must re-init each launch).
__global__ void ssd_init_ws(unsigned long long* __restrict__ ws, int n) {
  int i = blockIdx.x * blockDim.x + threadIdx.x;
  if (i < n) ws[i] = 0ull;
}

// K1: per-(batch, 512-prior chunk) block. Column maxima packed as
// (iou_bits << 32) | (0xFFFFFFFF - p)  -> max picks highest iou, smallest p on tie.
__global__ void __launch_bounds__(256)
ssd_colmax(const float4* __restrict__ gt,        // [B,G] corner boxes
           const float4* __restrict__ pri,       // [P] center-form priors
           unsigned long long* __restrict__ ws)  // [B,G] packed column max
{
  __shared__ float4 sGt[Gn];
  __shared__ float  sGA[Gn];
  __shared__ unsigned long long sBest[Gn];

  const int t = threadIdx.x;
  const int b = blockIdx.y;
  const int c = blockIdx.x;

  if ((t >> 5) == 0) {                                   // wave 0 issues the DMA
    tdm_load_gt_tile(gt + (size_t)b * Gn, (void*)sGt);
    __builtin_amdgcn_s_wait_tensorcnt(0);
  }
  if (t < Gn) sBest[t] = 0ull;
  __syncthreads();
  if (t < Gn) {
    float4 g4 = sGt[t];
    sGA[t] = fmaxf(g4.z - g4.x, 0.0f) * fmaxf(g4.w - g4.y, 0.0f);
  }
  __syncthreads();

  #pragma unroll
  for (int j = 0; j < 2; ++j) {
    int p = c * 512 + j * 256 + t;
    if (p < Pn) {
      float4 pc = pri[p];
      float hw = 0.5f * pc.z, hh = 0.5f * pc.w;
      float px1 = pc.x - hw, py1 = pc.y - hh, px2 = pc.x + hw, py2 = pc.y + hh;
      float pa = pc.z * pc.w;
      unsigned int pk = 0xFFFFFFFFu - (unsigned int)p;
      #pragma unroll 4
      for (int g = 0; g < Gn; ++g) {
        float iou = iou_one(px1, py1, px2, py2, pa, sGt[g], sGA[g]);
        unsigned long long packed =
            ((unsigned long long)__float_as_uint(iou) << 32) | (unsigned long long)pk;
        unsigned long long cur = *((volatile unsigned long long*)&sBest[g]);
        if (packed > cur) atomicMax(&sBest[g], packed);   // ds_max_u64
      }
    }
  }
  __syncthreads();

  if (t < Gn) {
    unsigned long long v = sBest[t];
    unsigned long long* dst = ws + (size_t)b * Gn + t;
    unsigned long long cur = *((volatile unsigned long long*)dst);
    if (v > cur) atomicMax(dst, v);                       // global_atomic_max_u64
  }
}

// K2: per-(batch, 256-prior chunk). Row argmax (recomputed), force-assign
// (ascending g, last wins -> matches scatter .set), encode, write.
__global__ void __launch_bounds__(256)
ssd_finalize(const float4* __restrict__ gt,
             const int* __restrict__ gl,                  // [B,G] labels (int32)
             const float4* __restrict__ pri,
             const unsigned long long* __restrict__ ws,   // [B,G] packed column max
             float* __restrict__ out)
{
  __shared__ float4 sGt[Gn];
  __shared__ float  sGA[Gn];
  __shared__ int    sLab[Gn];
  __shared__ int    sForced[Gn];

  const int t = threadIdx.x;
  const int b = blockIdx.y;
  const int p = blockIdx.x * 256 + t;

  if (p < Pn) __builtin_prefetch(&pri[p], 0, 0);          // global_prefetch_b8

  if ((t >> 5) == 0) {
    tdm_load_gt_tile(gt + (size_t)b * Gn, (void*)sGt);
    __builtin_amdgcn_s_wait_tensorcnt(0);
  }
  if (t < Gn) {
    sLab[t] = gl[b * Gn + t];
    sForced[t] = (int)(0xFFFFFFFFu -
                       (unsigned int)(ws[(size_t)b * Gn + t] & 0xFFFFFFFFull));
  }
  __syncthreads();
  if (t < Gn) {
    float4 g4 = sGt[t];
    sGA[t] = fmaxf(g4.z - g4.x, 0.0f) * fmaxf(g4.w - g4.y, 0.0f);
  }
  __syncthreads();

  if (p < Pn) {
    float4 pc = pri[p];
    float hw = 0.5f * pc.z, hh = 0.5f * pc.w;
    float px1 = pc.x - hw, py1 = pc.y - hh, px2 = pc.x + hw, py2 = pc.y + hh;
    float pa = pc.z * pc.w;

    float bv = -1.0f;
    int   bi = 0;
    #pragma unroll 4
    for (int g = 0; g < Gn; ++g) {
      float iou = iou_one(px1, py1, px2, py2, pa, sGt[g], sGA[g]);
      if (iou > bv) { bv = iou; bi = g; }                 // first max (jnp.argmax)
    }
    #pragma unroll 4
    for (int g = 0; g < Gn; ++g) {
      if (sForced[g] == p) { bi = g; bv = 2.0f; }         // last-wins force-assign
    }

    int lab = (bv < IOU_THRESHOLD) ? 0 : sLab[bi];
    float4 gb = sGt[bi];
    float cx = 0.5f * (gb.x + gb.z), cy = 0.5f * (gb.y + gb.w);
    float w  = gb.z - gb.x,          h  = gb.w - gb.y;

    float4 o;
    o.x = (cx - pc.x) / (pc.z * CENTER_VARIANCE);
    o.y = (cy - pc.y) / (pc.w * CENTER_VARIANCE);
    o.z = logf(w / pc.z) / SIZE_VARIANCE;
    o.w = logf(h / pc.w) / SIZE_VARIANCE;

    ((float4*)out)[(size_t)b * Pn + p] = o;
    out[(size_t)Bn * Pn * 4 + (size_t)b * Pn + p] = (float)lab;
  }
}

extern "C" void kernel_launch(void* const* d_in, const int* in_sizes, int n_in,
                              void* d_out, int out_size, void* d_ws, size_t ws_size,
                              hipStream_t stream) {
  const float4* gt  = (const float4*)d_in[0];   // [B,G,4] f32 corner boxes
  const int*    gl  = (const int*)d_in[1];      // [B,G] labels
  const float4* pri = (const float4*)d_in[2];   // [P,4] f32 center priors
  float* out = (float*)d_out;
  unsigned long long* ws = (unsigned long long*)d_ws;   // B*G*8 = 64 KB

  ssd_init_ws<<<(Bn * Gn + 255) / 256, 256, 0, stream>>>(ws, Bn * Gn);

  dim3 g1((Pn + 511) / 512, Bn);                // 18 x 128 = 2304 blocks
  ssd_colmax<<<g1, 256, 0, stream>>>(gt, pri, ws);

  dim3 g2((Pn + 255) / 256, Bn);                // 35 x 128 = 4480 blocks
  ssd_finalize<<<g2, 256, 0, stream>>>(gt, gl, pri, ws, out);
}